// FractalImage_80960133530081
// MI455X (gfx1250) — compile-verified
//
#include <hip/hip_runtime.h>
#include <hip/hip_bf16.h>
#include <math.h>

typedef __attribute__((ext_vector_type(2))) float v2f;
typedef __attribute__((ext_vector_type(8))) float v8f;

#define FI_BETA  4.0f
#define FI_E4    54.598150033144236f   // exp(4)
#define FI_VMAX  1.0e30f
#define FI_EPS   1.0e-8f
#define FI_B     16
#define FI_W     224
#define FI_NSEG  50
#define FI_RCAP  20
#define FI_ACCN  (FI_NSEG * FI_B * 3 * FI_RCAP)   // 48000 floats per accumulator array

// ---------------------------------------------------------------------------
// Kernel 0: zero the workspace accumulators (harness poisons ws with 0xAA).
// ---------------------------------------------------------------------------
__global__ void fi_zero(float* __restrict__ p, int n) {
    int i = blockIdx.x * 256 + threadIdx.x;
    if (i < n) p[i] = 0.0f;
}

// ---------------------------------------------------------------------------
// Kernel 1: per-patch soft box counting. One wave32 per patch, 8 waves/block.
// Patch pixels are staged global->LDS with async-to-LDS copies (double
// buffered, s_wait_asynccnt), freeing VGPRs for the 3x20 accumulators.
// ---------------------------------------------------------------------------
__global__ __launch_bounds__(256)
void fi_accum(const float* __restrict__ X,
              float* __restrict__ gn, float* __restrict__ gn2,
              int s, int k, int K2, int nw, int P, int R, int c,
              float rk, float rnw)
{
    __shared__ float sbuf[2][8][32];

    const int tid  = threadIdx.x;
    const int lane = tid & 31;
    const int w    = tid >> 5;
    const int b    = blockIdx.y;
    const int p    = blockIdx.x * 8 + w;
    if (p >= P) return;                       // uniform per wave

    // patch origin
    const int pi = (int)(((float)p + 0.5f) * rnw);
    const int pj = p - pi * nw;
    const float* src = X + ((size_t)b * FI_W + (size_t)pi * k) * FI_W + (size_t)pj * k;
    const float cen = src[c * FI_W + c];

    float acc[3][FI_RCAP];
#pragma unroll
    for (int m = 0; m < 3; ++m)
#pragma unroll
        for (int r = 0; r < FI_RCAP; ++r) acc[m][r] = 0.0f;

    const int T = (K2 + 31) >> 5;

    auto issue = [&](int t) {
        int e  = t * 32 + lane;
        int ec = (e < K2) ? e : (K2 - 1);      // clamp: every lane loads a valid addr
        int i  = (int)(((float)ec + 0.5f) * rk);
        int j  = ec - i * k;
        const float* gp = src + i * FI_W + j;
        unsigned lo = (unsigned)(size_t)&sbuf[t & 1][w][lane];
        unsigned long long ga = (unsigned long long)(size_t)gp;
        asm volatile("global_load_async_to_lds_b32 %0, %1, off"
                     :: "v"(lo), "v"(ga) : "memory");
    };

    issue(0);
    if (T > 1) issue(1);

    for (int t = 0; t < T; ++t) {
        if (t + 1 < T) asm volatile("s_wait_asynccnt 1" ::: "memory");
        else           asm volatile("s_wait_asynccnt 0" ::: "memory");

        float val = sbuf[t & 1][w][lane];
        asm volatile("s_wait_dscnt 0" ::: "memory");   // ds_read retired before buffer reuse
        if (t + 2 < T) issue(t + 2);

        int e  = t * 32 + lane;
        int ec = (e < K2) ? e : (K2 - 1);
        int i  = (int)(((float)ec + 0.5f) * rk);
        int j  = ec - i * k;
        float dy = fabsf((float)(i - c));
        float dx = fabsf((float)(j - c));
        float di = fabsf(val - cen);
        if (e >= K2) dy = 3.0e30f;             // duplicate lanes contribute exactly 0
        float dch = fmaxf(fmaxf(dy, dx), di);
        float deu = sqrtf(dy * dy + dx * dx + di * di);
        float dma = dy + dx + di;
        // v_r = exp(beta*(r-d)); sigmoid = v/(1+v); recurrence v *= e^beta
        float v0 = __expf(FI_BETA * (1.0f - dch));
        float v1 = __expf(FI_BETA * (1.0f - deu));
        float v2 = __expf(FI_BETA * (1.0f - dma));
#pragma unroll
        for (int r = 0; r < FI_RCAP; ++r) {
            if (r < R) {
                acc[0][r] += v0 * __builtin_amdgcn_rcpf(1.0f + v0);
                acc[1][r] += v1 * __builtin_amdgcn_rcpf(1.0f + v1);
                acc[2][r] += v2 * __builtin_amdgcn_rcpf(1.0f + v2);
                v0 = fminf(v0 * FI_E4, FI_VMAX);
                v1 = fminf(v1 * FI_E4, FI_VMAX);
                v2 = fminf(v2 * FI_E4, FI_VMAX);
            }
        }
    }

    // wave-level butterfly reduction of all (metric, radius) partial sums
#pragma unroll
    for (int m = 0; m < 3; ++m)
#pragma unroll
        for (int r = 0; r < FI_RCAP; ++r) {
            if (r < R) {
                float v = acc[m][r];
                v += __shfl_xor(v, 16, 32);
                v += __shfl_xor(v, 8, 32);
                v += __shfl_xor(v, 4, 32);
                v += __shfl_xor(v, 2, 32);
                v += __shfl_xor(v, 1, 32);
                acc[m][r] = v;
            }
        }

    if (lane == 0) {
        const float invK2 = 1.0f / (float)K2;
        const int base = ((s * FI_B + b) * 3) * FI_RCAP;
        for (int m = 0; m < 3; ++m)
            for (int r = 0; r < R; ++r) {
                float n = acc[m][r] * invK2;
                atomicAdd(&gn[base + m * FI_RCAP + r], n);
                atomicAdd(&gn2[base + m * FI_RCAP + r], n * n);
            }
    }
}

// ---------------------------------------------------------------------------
// Kernel 2: finalize. One wave handles 16 (size,batch,metric) triples.
// The fd log-log regression is a 16x20 * 20 mat-vec done with
// V_WMMA_F32_16X16X4_F32: xc weights folded into A, B = ones, 5 chained
// accumulating WMMAs -> D rows hold the dot products.
// ---------------------------------------------------------------------------
__device__ __forceinline__ void fi_decode(int q, int& s, int& b, int& mm,
                                          int& k, int& R, float& Pf) {
    s = q / 48;
    int rem = q - s * 48;
    b = rem / 3;
    mm = rem - b * 3;
    k = 3 + 2 * s;
    int nh = FI_W / k;
    Pf = (float)(nh * nh);
    R = (k - 1) / 2;
    if (R < 2) R = 2;
    if (R > FI_RCAP) R = FI_RCAP;
}

__global__ __launch_bounds__(32)
void fi_final(const float* __restrict__ gn, const float* __restrict__ gn2,
              float* __restrict__ out)
{
    const int lane  = threadIdx.x;
    const int qbase = blockIdx.x * 16;     // 150 blocks cover 2400 triples exactly
    const int row   = lane & 15;

    int s, b, mm, k, R;
    float Pf;
    fi_decode(qbase + row, s, b, mm, k, R, Pf);

    float lsum = 0.0f;
    for (int r = 0; r < R; ++r) lsum += logf((float)(r + 1));
    const float lxm = lsum / (float)R;
    const int base = ((s * FI_B + b) * 3 + mm) * FI_RCAP;

    float dot[16];   // only used by scalar fallback
    for (int i = 0; i < 16; ++i) dot[i] = 0.0f;

#if __has_builtin(__builtin_amdgcn_wmma_f32_16x16x4_f32)
    v8f cacc = {0.f, 0.f, 0.f, 0.f, 0.f, 0.f, 0.f, 0.f};
    v2f bones = {1.0f, 1.0f};
#pragma unroll
    for (int c5 = 0; c5 < 5; ++c5) {
        // A 16x4 f32 layout: VGPR0 = K (lane<16 ? 0 : 2), VGPR1 = K (lane<16 ? 1 : 3)
        int kx = 4 * c5 + (lane < 16 ? 0 : 2);
        int ky = kx + 1;
        v2f a;
        a.x = (kx < R) ? logf(gn[base + kx] / Pf + FI_EPS) * (logf((float)(kx + 1)) - lxm) : 0.0f;
        a.y = (ky < R) ? logf(gn[base + ky] / Pf + FI_EPS) * (logf((float)(ky + 1)) - lxm) : 0.0f;
        cacc = __builtin_amdgcn_wmma_f32_16x16x4_f32(false, a, false, bones,
                                                     (short)0, cacc, false, false);
    }
#endif

    // Writers: lane 0 owns D rows 0..7 (c[g] = row g), lane 16 owns rows 8..15.
    if (lane == 0 || lane == 16) {
        const int roff = (lane == 0) ? 0 : 8;
#pragma unroll
        for (int gg = 0; gg < 8; ++gg) {
            int s2, b2, m2, k2, R2;
            float P2;
            fi_decode(qbase + roff + gg, s2, b2, m2, k2, R2, P2);
            float ls = 0.0f;
            for (int r = 0; r < R2; ++r) ls += logf((float)(r + 1));
            float lxm2 = ls / (float)R2;
            float sxx = 0.0f;
            for (int r = 0; r < R2; ++r) {
                float xc = logf((float)(r + 1)) - lxm2;
                sxx += xc * xc;
            }
            const int b2ase = ((s2 * FI_B + b2) * 3 + m2) * FI_RCAP;
            float lac = 0.0f;
            float sdot = 0.0f;
            for (int r = 0; r < R2; ++r) {
                float m1  = gn[b2ase + r] / P2;
                float m2v = gn2[b2ase + r] / P2;
                lac += m2v / (m1 * m1 + FI_EPS);
                sdot += logf(m1 + FI_EPS) * (logf((float)(r + 1)) - lxm2);
            }
            lac /= (float)R2;
#if __has_builtin(__builtin_amdgcn_wmma_f32_16x16x4_f32)
            float d = cacc[gg];
#else
            float d = sdot; (void)dot;
#endif
            float fd = -d / sxx;
            out[b2 * 300 + s2 * 3 + m2] = lac;
            out[b2 * 300 + (50 + s2) * 3 + m2] = fd;
        }
    }
}

// ---------------------------------------------------------------------------
// Host launcher
// ---------------------------------------------------------------------------
extern "C" void kernel_launch(void* const* d_in, const int* in_sizes, int n_in,
                              void* d_out, int out_size, void* d_ws, size_t ws_size,
                              hipStream_t stream) {
    (void)in_sizes; (void)n_in; (void)out_size; (void)ws_size;
    const float* X = (const float*)d_in[0];
    float* out = (float*)d_out;
    float* gn  = (float*)d_ws;
    float* gn2 = gn + FI_ACCN;

    fi_zero<<<(2 * FI_ACCN + 255) / 256, 256, 0, stream>>>(gn, 2 * FI_ACCN);

    for (int si = 0; si < FI_NSEG; ++si) {
        int k  = 3 + 2 * si;
        int nh = FI_W / k;
        int nw = nh;
        int P  = nh * nw;
        int K2 = k * k;
        int R  = (k - 1) / 2;
        if (R < 2) R = 2;
        if (R > FI_RCAP) R = FI_RCAP;
        int c = k / 2;
        dim3 grid((P + 7) / 8, FI_B, 1);
        fi_accum<<<grid, 256, 0, stream>>>(X, gn, gn2, si, k, K2, nw, P, R, c,
                                           1.0f / (float)k, 1.0f / (float)nw);
    }

    fi_final<<<150, 32, 0, stream>>>(gn, gn2, out);
}